// HiddenMarkovGFormula_40295383171767
// MI455X (gfx1250) — compile-verified
//
#include <hip/hip_runtime.h>
#include <cstdint>
#include <cstddef>

#define T_STEPS 30
#define BLK 128   // 4 waves (wave32); 128 individuals per block

typedef float v2f __attribute__((ext_vector_type(2)));
typedef float v8f __attribute__((ext_vector_type(8)));

// ---- CDNA5 async global<->LDS DMA (ASYNCcnt-tracked) --------------------
__device__ __forceinline__ uint32_t lds_byte_off(const void* p) {
  // shared->generic addrspace cast places the LDS offset in the low 32 bits
  return (uint32_t)(uintptr_t)p;
}
__device__ __forceinline__ void async_load_b128(uint32_t lds_off, const float* gptr) {
  asm volatile("global_load_async_to_lds_b128 %0, %1, off"
               :: "v"(lds_off), "v"((uint64_t)(uintptr_t)gptr)
               : "memory");
}
__device__ __forceinline__ void async_store_b128(float* gptr, uint32_t lds_off) {
  asm volatile("global_store_async_from_lds_b128 %0, %1, off"
               :: "v"((uint64_t)(uintptr_t)gptr), "v"(lds_off)
               : "memory");
}
__device__ __forceinline__ void wait_async0() {
  asm volatile("s_wait_asynccnt 0" ::: "memory");
}
__device__ __forceinline__ float lane_bcast_f(float v, int srcLane) {
  return __int_as_float(__builtin_amdgcn_ds_bpermute(srcLane << 2, __float_as_int(v)));
}
// single v_rcp_f32 (~1 ulp) instead of a full-precision divide sequence
__device__ __forceinline__ float fast_rcp(float x) { return __builtin_amdgcn_rcpf(x); }

__global__ __launch_bounds__(BLK)
void hmm_filter_kernel(const float* __restrict__ G,  const float* __restrict__ S,
                       const float* __restrict__ C,  const float* __restrict__ Y,
                       const float* __restrict__ L,
                       const float* p_psi, const float* p_gS, const float* p_gC,
                       const float* p_gG, const float* p_gGS, const float* p_gGC,
                       const float* p_gLw, const float* p_lsZ,
                       const float* p_b0, const float* p_bZ, const float* p_bS,
                       const float* p_bG, const float* p_bGS, const float* p_bLw,
                       const float* p_bt,
                       float* __restrict__ outZf, float* __restrict__ outZv,
                       float* __restrict__ partials, int N)
{
  __shared__ __align__(16) float sS[BLK * T_STEPS];
  __shared__ __align__(16) float sC[BLK * T_STEPS];
  __shared__ __align__(16) float sY[BLK * T_STEPS];
  __shared__ __align__(16) float sZm[BLK * T_STEPS];
  __shared__ __align__(16) float sZv[BLK * T_STEPS];
  __shared__ float sRed[BLK / 32];

  const int  tid       = threadIdx.x;
  const int  blockBase = blockIdx.x * BLK;
  const bool fullBlock = (blockBase + BLK) <= N;
  const int  tileElems = BLK * T_STEPS;                 // 3840 floats / array
  const size_t gElem   = (size_t)blockBase * T_STEPS;   // element offset of tile

  // ---- stage S/C/Y tile into LDS -------------------------------------------
  if (fullBlock) {
    const uint32_t oS = lds_byte_off(sS), oC = lds_byte_off(sC), oY = lds_byte_off(sY);
    for (int i = tid; i < tileElems / 4; i += BLK) {    // 960 x b128 per array
      const uint32_t lo = (uint32_t)i * 16u;
      async_load_b128(oS + lo, S + gElem + (size_t)i * 4);
      async_load_b128(oC + lo, C + gElem + (size_t)i * 4);
      async_load_b128(oY + lo, Y + gElem + (size_t)i * 4);
    }
    wait_async0();
  } else {
    for (int i = tid; i < tileElems; i += BLK) { sS[i] = 0.f; sC[i] = 0.f; sY[i] = 0.f; }
    const int validElems = (N - blockBase) * T_STEPS;   // same thread re-touches same slots
    for (int i = tid; i < validElems; i += BLK) {
      const size_t g = gElem + (size_t)i;
      sS[i] = S[g]; sC[i] = C[g]; sY[i] = Y[g];
    }
  }
  __syncthreads();

  // ---- uniform scalar parameters ------------------------------------------
  const float psi  = p_psi[0];
  const float gS_  = p_gS[0],  gC_  = p_gC[0], gG_ = p_gG[0];
  const float gGS_ = p_gGS[0], gGC_ = p_gGC[0];
  const float b0_  = p_b0[0],  bZ_  = p_bZ[0], bS_ = p_bS[0];
  const float bG_  = p_bG[0],  bGS_ = p_bGS[0], bt_ = p_bt[0];
  const float sigma2 = __expf(2.0f * p_lsZ[0]);         // exp(lsZ)^2
  const float gw0 = p_gLw[0], gw1 = p_gLw[1], gw2 = p_gLw[2], gw3 = p_gLw[3], gw4 = p_gLw[4];
  const float bw0 = p_bLw[0], bw1 = p_bLw[1], bw2 = p_bLw[2], bw3 = p_bLw[3], bw4 = p_bLw[4];

  // ---- per-thread individual (clamped so EXEC stays all-ones for WMMA) ----
  const int n  = blockBase + tid;
  const int nc = (n < N) ? n : (N - 1);
  const float l0 = L[(size_t)nc * 5 + 0];
  const float l1 = L[(size_t)nc * 5 + 1];
  const float l2 = L[(size_t)nc * 5 + 2];
  const float l3 = L[(size_t)nc * 5 + 3];
  const float l4 = L[(size_t)nc * 5 + 4];

  // ---- gL = L.gamma, bL = L.beta via V_WMMA_F32_16X16X4_F32 ----------------
  // B[k][col] = L[wave individual col][k]; A rows all = weight vector, so every
  // D column equals the per-individual dot product over k=0..3. k=4 is an FMA.
  const int  lane = tid & 31;
  const bool hiH  = lane >= 16;
  const int  srcA = lane & 15;          // wave individuals 0..15
  const int  srcB = 16 + (lane & 15);   // wave individuals 16..31

  v2f bMatA, bMatB;
  {
    const float a0 = lane_bcast_f(l0, srcA), a1 = lane_bcast_f(l1, srcA);
    const float a2 = lane_bcast_f(l2, srcA), a3 = lane_bcast_f(l3, srcA);
    bMatA.x = hiH ? a2 : a0;            // VGPR0: K = 0 (lanes 0-15) / 2 (16-31)
    bMatA.y = hiH ? a3 : a1;            // VGPR1: K = 1 / 3
    const float c0 = lane_bcast_f(l0, srcB), c1 = lane_bcast_f(l1, srcB);
    const float c2 = lane_bcast_f(l2, srcB), c3 = lane_bcast_f(l3, srcB);
    bMatB.x = hiH ? c2 : c0;
    bMatB.y = hiH ? c3 : c1;
  }
  v2f aGam; aGam.x = hiH ? gw2 : gw0; aGam.y = hiH ? gw3 : gw1;
  v2f aBet; aBet.x = hiH ? bw2 : bw0; aBet.y = hiH ? bw3 : bw1;

  const v8f zero8 = {};
  v8f dGA = __builtin_amdgcn_wmma_f32_16x16x4_f32(false, aGam, false, bMatA, (short)0, zero8, false, false);
  v8f dGB = __builtin_amdgcn_wmma_f32_16x16x4_f32(false, aGam, false, bMatB, (short)0, zero8, false, false);
  v8f dBA = __builtin_amdgcn_wmma_f32_16x16x4_f32(false, aBet, false, bMatA, (short)0, zero8, false, false);
  v8f dBB = __builtin_amdgcn_wmma_f32_16x16x4_f32(false, aBet, false, bMatB, (short)0, zero8, false, false);

  const float gL = (hiH ? dGB[0] : dGA[0]) + l4 * gw4;  // D[0][lane] / D[8][lane-16]
  const float bL = (hiH ? dBB[0] : dBA[0]) + l4 * bw4;

  // ---- sequential filter over T=30 ----------------------------------------
  const float Gn   = G[nc];
  const float cZ   = gG_ * Gn + gL;          // time-invariant transition term
  const float aS   = gS_ + gGS_ * Gn;
  const float aC   = gC_ + gGC_ * Gn;
  const float lb   = b0_ + bG_ * Gn + bL;    // time-invariant logit term
  const float lSc  = bS_ + bGS_ * Gn;
  const float psi2 = psi * psi;
  const float bZ2  = bZ_ * bZ_;

  float Zm = 0.f, Zv = 1.f, ll = 0.f;
  const int rowOff = tid * T_STEPS;          // stride 30 across lanes: conflict-free
#pragma unroll
  for (int t = 0; t < T_STEPS; ++t) {
    const float St = sS[rowOff + t];
    const float Ct = sC[rowOff + t];
    const float Yt = sY[rowOff + t];

    const float Zpm = fmaf(psi, Zm, fmaf(aS, St, fmaf(aC, Ct, cZ)));
    const float Zpv = fmaf(psi2, Zv, sigma2);

    float logit = fmaf(bZ_, Zpm, fmaf(lSc, St, lb + bt_ * (float)t));
    logit = fminf(20.f, fmaxf(-20.f, logit));

    // p = sigmoid(logit) via one v_exp + one v_rcp
    const float e = __expf(-logit);
    const float p = fast_rcp(1.f + e);

    const float grad = (Yt - p) * bZ_;
    const float hess = fmaf(bZ2, p * (1.f - p), 1e-6f);
    // 1/(1/Zpv + hess) == Zpv / (1 + Zpv*hess)  -> single v_rcp
    const float Zpost_v = Zpv * fast_rcp(fmaf(Zpv, hess, 1.f));
    const float Zpost_m = fmaf(Zpost_v, grad, Zpm);

    ll += fmaf(Yt, __logf(p + 1e-10f), (1.f - Yt) * __logf(1.f - p + 1e-10f));

    sZm[rowOff + t] = Zpost_m;
    sZv[rowOff + t] = Zpost_v;
    Zm = Zpost_m; Zv = Zpost_v;
  }
  if (n >= N) ll = 0.f;

  // ---- deterministic block reduction of ll --------------------------------
  for (int o = 16; o > 0; o >>= 1) ll += __shfl_down(ll, o, 32);
  if ((tid & 31) == 0) sRed[tid >> 5] = ll;
  __syncthreads();                            // also fences sZm/sZv for store-out
  if (tid == 0) {
    float s = 0.f;
    for (int w = 0; w < BLK / 32; ++w) s += sRed[w];
    partials[blockIdx.x] = s;
  }

  // ---- stage Z_filtered / Z_variance back out ------------------------------
  if (fullBlock) {
    const uint32_t oM = lds_byte_off(sZm), oV = lds_byte_off(sZv);
    for (int i = tid; i < tileElems / 4; i += BLK) {
      const uint32_t lo = (uint32_t)i * 16u;
      async_store_b128(outZf + gElem + (size_t)i * 4, oM + lo);
      async_store_b128(outZv + gElem + (size_t)i * 4, oV + lo);
    }
    wait_async0();
  } else {
    const int validElems = (N - blockBase) * T_STEPS;
    for (int i = tid; i < validElems; i += BLK) {
      const size_t g = gElem + (size_t)i;
      outZf[g] = sZm[i];
      outZv[g] = sZv[i];
    }
  }
}

// Fixed-order final reduction -> out[2*N*T] (bitwise deterministic across replays)
__global__ __launch_bounds__(256)
void ll_reduce_kernel(const float* __restrict__ partials, int nPart, float* __restrict__ outLL)
{
  __shared__ float red[256];
  float s = 0.f;
  for (int i = threadIdx.x; i < nPart; i += 256) s += partials[i];
  red[threadIdx.x] = s;
  __syncthreads();
  for (int o = 128; o > 0; o >>= 1) {
    if ((int)threadIdx.x < o) red[threadIdx.x] += red[threadIdx.x + o];
    __syncthreads();
  }
  if (threadIdx.x == 0) *outLL = red[0];
}

extern "C" void kernel_launch(void* const* d_in, const int* in_sizes, int n_in,
                              void* d_out, int out_size, void* d_ws, size_t ws_size,
                              hipStream_t stream) {
  const float* G   = (const float*)d_in[0];
  const float* S   = (const float*)d_in[1];
  const float* C   = (const float*)d_in[2];
  const float* Y   = (const float*)d_in[3];
  const float* L   = (const float*)d_in[4];
  const float* psi = (const float*)d_in[5];
  const float* gS  = (const float*)d_in[6];
  const float* gC  = (const float*)d_in[7];
  const float* gG  = (const float*)d_in[8];
  const float* gGS = (const float*)d_in[9];
  const float* gGC = (const float*)d_in[10];
  const float* gLw = (const float*)d_in[11];
  const float* lsZ = (const float*)d_in[12];
  const float* b0  = (const float*)d_in[13];
  const float* bZ  = (const float*)d_in[14];
  const float* bS  = (const float*)d_in[15];
  const float* bG  = (const float*)d_in[16];
  const float* bGS = (const float*)d_in[17];
  const float* bLw = (const float*)d_in[18];
  const float* bt  = (const float*)d_in[19];

  const int    N  = in_sizes[0];
  const size_t NT = (size_t)N * T_STEPS;

  float* out      = (float*)d_out;
  float* outZf    = out;            // [N*T]
  float* outZv    = out + NT;       // [N*T]
  float* outLL    = out + 2 * NT;   // [1]
  float* partials = (float*)d_ws;   // nBlocks floats

  const int nBlocks = (N + BLK - 1) / BLK;

  hmm_filter_kernel<<<nBlocks, BLK, 0, stream>>>(
      G, S, C, Y, L, psi, gS, gC, gG, gGS, gGC, gLw, lsZ,
      b0, bZ, bS, bG, bGS, bLw, bt, outZf, outZv, partials, N);

  ll_reduce_kernel<<<1, 256, 0, stream>>>(partials, nBlocks, outLL);
}